// none_local_block_88957362634884
// MI455X (gfx1250) — compile-verified
//
#include <hip/hip_runtime.h>
#include <math.h>

// Non-local block: B=4, H=W=64, C=128, D=64, N=4096
#define Bb 4
#define Cc 128
#define Dd 64
#define Nn 4096

typedef float v2f __attribute__((ext_vector_type(2)));
typedef float v8f __attribute__((ext_vector_type(8)));
typedef unsigned int u32x4 __attribute__((ext_vector_type(4)));
typedef int i32x4 __attribute__((ext_vector_type(4)));
typedef int i32x8 __attribute__((ext_vector_type(8)));

static __device__ __forceinline__ v8f wmma_f32(v2f a, v2f b, v8f c) {
  // D = A(16x4,f32) * B(4x16,f32) + C(16x16,f32)
  return __builtin_amdgcn_wmma_f32_16x16x4_f32(
      /*neg_a=*/false, a, /*neg_b=*/false, b,
      /*c_mod=*/(short)0, c, /*reuse_a=*/false, /*reuse_b=*/false);
}

// TDM: DMA one 16-row x 64-col f32 tile (row stride 64 floats) into LDS.
// D# layout per CDNA5 ISA (2D tensor; higher-dim
// descriptor groups zero-filled). 6-arg builtin form for this toolchain.
static __device__ __forceinline__ void tdm_load_tile_16x64(const float* gptr,
                                                           unsigned lds_off) {
  unsigned long long ga = (unsigned long long)gptr;
  u32x4 g0;
  g0.x = 0x1u;                                   // count=1 (valid user desc)
  g0.y = lds_off;                                // lds_addr (bytes)
  g0.z = (unsigned)ga;                           // global_addr[31:0]
  g0.w = (unsigned)((ga >> 32) & 0x1FFFFFFu)     // global_addr[56:32]
         | (2u << 30);                           // type=2 ("image")
  i32x8 g1;
  g1[0] = 0x00020000;                            // data_size=2 (4 bytes)
  g1[1] = (int)(64u << 16);                      // tensor_dim0 = 64 (lo16)
  g1[2] = (int)(4096u << 16);                    // dim0 hi=0 | tensor_dim1 lo=4096
  g1[3] = (int)(64u << 16);                      // dim1 hi=0 | tile_dim0 = 64
  g1[4] = 16;                                    // tile_dim1=16, tile_dim2=0
  g1[5] = 64;                                    // tensor_dim0_stride = 64 (lo32)
  g1[6] = 0;                                     // stride hi | dim1_stride lo
  g1[7] = 0;
  i32x4 z4 = {0, 0, 0, 0};
  i32x8 z8 = {0, 0, 0, 0, 0, 0, 0, 0};
  __builtin_amdgcn_tensor_load_to_lds(g0, g1, z4, z4, z8, 0);
}

// ---------------------------------------------------------------------------
// Kernel 1: fused k/q/v projections.  y = X @ W,  X:[B*N,C], W:[C,D]
// ---------------------------------------------------------------------------
__global__ __launch_bounds__(256) void proj_kernel(
    const float* __restrict__ x0, const float* __restrict__ x1,
    const float* __restrict__ Wk, const float* __restrict__ Wq,
    const float* __restrict__ Wv,
    float* __restrict__ kbuf, float* __restrict__ qbuf,
    float* __restrict__ vbuf) {
  const int lane = threadIdx.x & 31;
  const int wave = threadIdx.x >> 5;
  const int gw = blockIdx.x * 8 + wave;      // 0 .. 3*1024-1
  const int m  = gw >> 10;                   // 0=k, 1=q, 2=v
  const int t  = gw & 1023;                  // row tile
  const int row0 = t * 16;

  const float* X = (m == 2) ? x1 : x0;
  const float* W = (m == 0) ? Wk : ((m == 1) ? Wq : Wv);
  float*       O = (m == 0) ? kbuf : ((m == 1) ? qbuf : vbuf);

  const int rc = lane & 15;                  // A-row / B-col / C-col
  const int kb = (lane >> 4) * 2;            // K sub-base within 4-chunk
  const int h8 = (lane >> 4) * 8;            // C/D row offset for upper half

  v8f acc[4];
#pragma unroll
  for (int ct = 0; ct < 4; ++ct) acc[ct] = (v8f){};

  const float* xrow = X + (size_t)(row0 + rc) * Cc;
#pragma unroll 4
  for (int kk = 0; kk < Cc; kk += 4) {
    v2f a = *(const v2f*)(xrow + kk + kb);   // A[rc][kk+kb .. +1]
#pragma unroll
    for (int ct = 0; ct < 4; ++ct) {
      const float* wp = W + (size_t)(kk + kb) * Dd + ct * 16 + rc;
      v2f b;
      b.x = wp[0];                           // B[kb  ][col]
      b.y = wp[Dd];                          // B[kb+1][col]
      acc[ct] = wmma_f32(a, b, acc[ct]);
    }
  }
#pragma unroll
  for (int ct = 0; ct < 4; ++ct)
#pragma unroll
    for (int r = 0; r < 8; ++r)
      O[(size_t)(row0 + r + h8) * Dd + ct * 16 + rc] = acc[ct][r];
}

// ---------------------------------------------------------------------------
// Kernel 2: flash attention + fused output projection, TDM double-buffered.
// scores[i,j] = k_i . q_j ; softmax over j ; out_i = sum_j attn * v_j ; @Wout
// Block = 8 waves, each owns one 16-row i-tile; Q_j/V_j tiles are DMA'd to
// LDS once per block by the Tensor Data Mover and shared by all 8 waves.
// ---------------------------------------------------------------------------
__global__ __launch_bounds__(256) void attn_kernel(
    const float* __restrict__ kbuf, const float* __restrict__ qbuf,
    const float* __restrict__ vbuf, const float* __restrict__ Wout,
    float* __restrict__ out) {
  __shared__ float qtile[2][16 * 64];        // double-buffered Q_j tile
  __shared__ float vtile[2][16 * 64];        // double-buffered V_j tile
  __shared__ float pscr[8][16 * 64];         // per-wave P/O staging

  const int lane = threadIdx.x & 31;
  const int wave = threadIdx.x >> 5;
  float* wl = &pscr[wave][0];

  const int gw = blockIdx.x * 8 + wave;      // 0 .. 1023
  const int bb = gw >> 8;                    // batch (uniform per block)
  const int i0 = (gw & 255) * 16;            // i-tile start

  const int rc = lane & 15;
  const int kb = (lane >> 4) * 2;
  const int h8 = (lane >> 4) * 8;

  const float* Kp = kbuf + (size_t)bb * Nn * Dd;
  const float* Qp = qbuf + (size_t)bb * Nn * Dd;
  const float* Vp = vbuf + (size_t)bb * Nn * Dd;

  // Pin K_i (16x64) in registers as 16 A-fragments (K-dim chunks of 4)
  v2f Ka[16];
#pragma unroll
  for (int dk = 0; dk < 16; ++dk)
    Ka[dk] = *(const v2f*)(Kp + (size_t)(i0 + rc) * Dd + dk * 4 + kb);

  v8f o[4];
#pragma unroll
  for (int ct = 0; ct < 4; ++ct) o[ct] = (v8f){};
  float mrow[8], lrow[8];
#pragma unroll
  for (int r = 0; r < 8; ++r) { mrow[r] = -3.0e38f; lrow[r] = 0.0f; }

  // Prologue: DMA chunks 0 and 1 into the two buffers.
  if (wave == 0) {
    tdm_load_tile_16x64(Qp,      (unsigned)(unsigned long long)&qtile[0][0]);
    tdm_load_tile_16x64(Vp,      (unsigned)(unsigned long long)&vtile[0][0]);
    tdm_load_tile_16x64(Qp + 16 * Dd, (unsigned)(unsigned long long)&qtile[1][0]);
    tdm_load_tile_16x64(Vp + 16 * Dd, (unsigned)(unsigned long long)&vtile[1][0]);
  }

  for (int it = 0; it < Nn / 16; ++it) {
    const int cur = it & 1;
    if (wave == 0) {
      // TDM ops complete in order: leaving the 2 newest (chunk it+1)
      // outstanding still guarantees chunk `it` has landed.
      if (it + 1 < Nn / 16) __builtin_amdgcn_s_wait_tensorcnt(2);
      else                  __builtin_amdgcn_s_wait_tensorcnt(0);
    }
    __syncthreads();                          // buf[cur] ready for all waves

    const float* qt = &qtile[cur][0];
    const float* vt = &vtile[cur][0];

    // S = K_i (16x64) @ Q_j^T (64x16), B-fragments from LDS
    v8f s = (v8f){};
#pragma unroll
    for (int dk = 0; dk < 16; ++dk) {
      v2f b = *(const v2f*)(qt + rc * 64 + dk * 4 + kb);  // B[k][n]=Q[j0+n][k]
      s = wmma_f32(Ka[dk], b, s);
    }
    // online softmax per accumulator row (rows split across half-waves)
#pragma unroll
    for (int r = 0; r < 8; ++r) {
      float v = s[r];
      float mx = v;
      mx = fmaxf(mx, __shfl_xor(mx, 1, 16));
      mx = fmaxf(mx, __shfl_xor(mx, 2, 16));
      mx = fmaxf(mx, __shfl_xor(mx, 4, 16));
      mx = fmaxf(mx, __shfl_xor(mx, 8, 16));
      float mnew  = fmaxf(mrow[r], mx);
      float scale = __expf(mrow[r] - mnew);
      float p     = __expf(v - mnew);
      float sum = p;
      sum += __shfl_xor(sum, 1, 16);
      sum += __shfl_xor(sum, 2, 16);
      sum += __shfl_xor(sum, 4, 16);
      sum += __shfl_xor(sum, 8, 16);
      lrow[r] = lrow[r] * scale + sum;
      mrow[r] = mnew;
#pragma unroll
      for (int ct = 0; ct < 4; ++ct) o[ct][r] *= scale;
      wl[(r + h8) * 16 + rc] = p;            // stash P in D-layout
    }
    __builtin_amdgcn_wave_barrier();         // order LDS store -> load
    // O += P (16x16) @ V_j (16x64), V B-fragments from LDS
#pragma unroll
    for (int kc = 0; kc < 4; ++kc) {
      v2f a = *(const v2f*)(wl + rc * 16 + kc * 4 + kb);  // P as A-fragment
      const float* vrow = vt + (kc * 4 + kb) * 64;
#pragma unroll
      for (int ct = 0; ct < 4; ++ct) {
        v2f b;
        b.x = vrow[ct * 16 + rc];
        b.y = vrow[64 + ct * 16 + rc];
        o[ct] = wmma_f32(a, b, o[ct]);
      }
    }
    __syncthreads();                          // all waves done with buf[cur]
    if (wave == 0 && it + 2 < Nn / 16) {      // refill the freed buffer
      const float* qn = Qp + (size_t)(it + 2) * 16 * Dd;
      const float* vn = Vp + (size_t)(it + 2) * 16 * Dd;
      tdm_load_tile_16x64(qn, (unsigned)(unsigned long long)&qtile[cur][0]);
      tdm_load_tile_16x64(vn, (unsigned)(unsigned long long)&vtile[cur][0]);
    }
  }

  // Normalize and stage O (16x64) in per-wave LDS for the Wout projection
#pragma unroll
  for (int r = 0; r < 8; ++r) {
    float inv = 1.0f / lrow[r];
#pragma unroll
    for (int ct = 0; ct < 4; ++ct)
      wl[(r + h8) * 64 + ct * 16 + rc] = o[ct][r] * inv;
  }
  __builtin_amdgcn_wave_barrier();

  // final = O (16x64) @ Wout (64x128), written straight to [B,N,C]
#pragma unroll
  for (int ct2 = 0; ct2 < 8; ++ct2) {
    v8f f = (v8f){};
#pragma unroll
    for (int dk = 0; dk < 16; ++dk) {
      v2f a = *(const v2f*)(wl + rc * 64 + dk * 4 + kb);
      const float* wp = Wout + (size_t)(dk * 4 + kb) * Cc + ct2 * 16 + rc;
      v2f b;
      b.x = wp[0];
      b.y = wp[Cc];
      f = wmma_f32(a, b, f);
    }
#pragma unroll
    for (int r = 0; r < 8; ++r)
      out[(size_t)((bb * Nn) + i0 + r + h8) * Cc + ct2 * 16 + rc] = f[r];
  }
}

// ---------------------------------------------------------------------------
extern "C" void kernel_launch(void* const* d_in, const int* in_sizes, int n_in,
                              void* d_out, int out_size, void* d_ws,
                              size_t ws_size, hipStream_t stream) {
  const float* x0   = (const float*)d_in[0];
  const float* x1   = (const float*)d_in[1];
  const float* Wk   = (const float*)d_in[2];
  const float* Wq   = (const float*)d_in[3];
  const float* Wv   = (const float*)d_in[4];
  const float* Wout = (const float*)d_in[5];
  float* out = (float*)d_out;

  const size_t kqv = (size_t)Bb * Nn * Dd;   // 1M floats = 4 MB each
  float* kbuf = (float*)d_ws;
  float* qbuf = kbuf + kqv;
  float* vbuf = qbuf + kqv;

  // 3 matrices * (B*N/16)=1024 row-tiles, 8 waves/block
  proj_kernel<<<(3 * 1024) / 8, 256, 0, stream>>>(x0, x1, Wk, Wq, Wv,
                                                  kbuf, qbuf, vbuf);
  // B*(N/16)=1024 i-tiles, 8 waves/block
  attn_kernel<<<1024 / 8, 256, 0, stream>>>(kbuf, qbuf, vbuf, Wout, out);
}